// BilinearGeneral_20942260535354
// MI455X (gfx1250) — compile-verified
//
#include <hip/hip_runtime.h>

#define BATCH 1024
#define IN1D  512
#define IN2D  512
#define OUTD  512
#define KTOT  (IN1D * IN2D)                 // 262144 contraction length
#define KSPLIT 16
#define I_PER_WG ((KTOT / KSPLIT) / IN2D)   // 32 "i" rows per workgroup
#define BM 256                              // batch tile per WG
#define BN 64                               // out tile per WG (2 wave cols * 32)
#define ZPITCH 520                          // 260 dwords/row -> conflict-free b128

typedef _Float16 h8   __attribute__((ext_vector_type(8)));
typedef _Float16 h16  __attribute__((ext_vector_type(16)));
typedef float    v8f  __attribute__((ext_vector_type(8)));
typedef float    f4   __attribute__((ext_vector_type(4)));
typedef float    f8v  __attribute__((ext_vector_type(8)));
typedef float    f16v __attribute__((ext_vector_type(16)));

__device__ __forceinline__ h8 wmma_f16d(h16 a, h16 b, h8 c) {
    // D(f16) = A(16x32 f16) * B(32x16 f16) + C(16x16 f16)  -- 4-VGPR C/D
    return __builtin_amdgcn_wmma_f16_16x16x32_f16(
        false, a, false, b, (short)0, c, false, false);
}

// ---------------------------------------------------------------------------
// Kernel 1: out[b,k] = bias[k] + z[b,:]·U[k,:] + x[b,:]·V[k,:]  (0.4% of work)
// Also initializes d_out for the atomic accumulation of the bilinear term.
// ---------------------------------------------------------------------------
__global__ __launch_bounds__(256) void bilinear_init_uvb(
    const float* __restrict__ x, const float* __restrict__ z,
    const float* __restrict__ U, const float* __restrict__ V,
    const float* __restrict__ bias, float* __restrict__ out)
{
    __shared__ float xs[IN1D];
    __shared__ float zs[IN2D];
    const int b = blockIdx.x;
    const int t = threadIdx.x;
    for (int c = t; c < IN1D; c += 256) {
        xs[c] = x[(size_t)b * IN1D + c];
        zs[c] = z[(size_t)b * IN2D + c];
    }
    __syncthreads();
    const int k = blockIdx.y * 256 + t;
    const float* Uk = U + (size_t)k * IN2D;
    const float* Vk = V + (size_t)k * IN1D;
    float s = bias[k];
    #pragma unroll 8
    for (int j = 0; j < IN2D; ++j) s += zs[j] * Uk[j];
    #pragma unroll 8
    for (int i = 0; i < IN1D; ++i) s += xs[i] * Vk[i];
    out[(size_t)b * OUTD + k] = s;
}

// ---------------------------------------------------------------------------
// Kernel 2: fused bilinear GEMM via WMMA, x-scaling deferred to accumulator:
//   out[b,k] = sum_i x[b,i] * ( sum_j z[b,j] * W[k,i,j] )
// Per i: WMMA plain-z A-frags into an f16 tacc (4 VGPRs/frag) over the 512-
// wide j sweep, then acc(f32) += x[:,i] (*) tacc.  No barriers / LDS stores
// in the K loop.  BM=256 -> W read 4x per element: 2 GB L2, 512 MB HBM.
// ---------------------------------------------------------------------------
__global__ __launch_bounds__(256) void bilinear_wmma(
    const float* __restrict__ x, const float* __restrict__ z,
    const float* __restrict__ W, float* __restrict__ out)
{
    __shared__ _Float16 zsh[BM][ZPITCH];       // 260 KB: z rows as f16 (static)
    __shared__ float    xsh_t[I_PER_WG][BM];   //  32 KB: x transposed (static)

    const int tid  = threadIdx.x;
    const int lane = tid & 31;
    const int wid  = tid >> 5;
    const int l15  = lane & 15;
    const int kh   = lane >> 4;              // K-half select (wave32 WMMA layout)

    const int bbase  = blockIdx.x * BM;                  // batch (M) base of WG
    const int mwbase = (wid >> 1) * 64;                  // wave M sub-tile base
    const int ngbase = blockIdx.y * BN + (wid & 1) * 32; // out (N) base of wave
    const int ibase  = blockIdx.z * I_PER_WG;            // K-split base (i units)

    // ---- one-time staging (then the K loop is barrier-free) ----------------
    for (int c = tid; c < BM * (IN2D / 8); c += 256) {
        const int r   = c >> 6;
        const int col = (c & 63) * 8;
        f8v zf = *(const f8v*)(z + (size_t)(bbase + r) * IN2D + col);
        *(h8*)&zsh[r][col] = __builtin_convertvector(zf, h8);
    }
    {   // x panel transposed: xsh_t[ii][r] = x[bbase+r][ibase+ii], f32
        const int r = tid;                   // one row per thread (BM == 256)
        const float* xp = x + (size_t)(bbase + r) * IN1D + ibase;
        #pragma unroll
        for (int q = 0; q < I_PER_WG / 4; ++q) {
            f4 xv = *(const f4*)(xp + q * 4);
            #pragma unroll
            for (int t = 0; t < 4; ++t) xsh_t[q * 4 + t][r] = xv[t];
        }
    }
    __syncthreads();

    v8f acc[4][2] = {};

    // Per-lane W row pointers (GEMM "B" operand, K contiguous in memory).
    const float* wbase0 = W + (size_t)(ngbase + l15) * KTOT + (size_t)kh * 16;
    const float* wbase1 = wbase0 + (size_t)16 * KTOT;

    // A-fragment from static z panel: two 8-half runs (ISA 16-bit A layout).
    auto load_a = [&](int mt, int j0) -> h16 {
        const _Float16* arow = &zsh[mwbase + mt * 16 + l15][0];
        h8 lo = *(const h8*)(arow + j0 + 8 * kh);
        h8 hi = *(const h8*)(arow + j0 + 16 + 8 * kh);
        return __builtin_shufflevector(lo, hi,
               0,1,2,3,4,5,6,7,8,9,10,11,12,13,14,15);
    };

    for (int ii = 0; ii < I_PER_WG; ++ii) {
        const int i = ibase + ii;
        const float* w0 = wbase0 + (size_t)i * IN2D;
        const float* w1 = wbase1 + (size_t)i * IN2D;

        h8 tacc[4][2];                       // f16 partials: 4 VGPRs each

        {   // jt = 0 peeled: C operand is constant zero (inline 0 in v_wmma)
            const h8 zero = {};
            h16 b0 = __builtin_convertvector(*(const f16v*)(w0), h16);
            h16 b1 = __builtin_convertvector(*(const f16v*)(w1), h16);
            #pragma unroll
            for (int mt = 0; mt < 4; ++mt) {
                h16 a = load_a(mt, 0);
                tacc[mt][0] = wmma_f16d(a, b0, zero);
                tacc[mt][1] = wmma_f16d(a, b1, zero);
            }
        }

        #pragma unroll 1
        for (int jt = 1; jt < IN2D / 32; ++jt) {
            const int j0 = jt * 32;
            __builtin_prefetch(w0 + j0 + 128, 0, 1);   // global_prefetch_b8
            __builtin_prefetch(w1 + j0 + 128, 0, 1);

            h16 b0 = __builtin_convertvector(*(const f16v*)(w0 + j0), h16);
            h16 b1 = __builtin_convertvector(*(const f16v*)(w1 + j0), h16);
            #pragma unroll
            for (int mt = 0; mt < 4; ++mt) {
                h16 a = load_a(mt, j0);
                tacc[mt][0] = wmma_f16d(a, b0, tacc[mt][0]);
                tacc[mt][1] = wmma_f16d(a, b1, tacc[mt][1]);
            }
        }

        // acc(f32) += x[:, i] (*) tacc ; x-frag is an LDS broadcast read.
        // f16 C/D layout: element e <-> M = e + 8*kh (same mapping as f32).
        #pragma unroll
        for (int mt = 0; mt < 4; ++mt) {
            f8v xv = *(const f8v*)&xsh_t[ii][mwbase + mt * 16 + 8 * kh];
            acc[mt][0] += __builtin_convertvector(tacc[mt][0], f8v) * xv;
            acc[mt][1] += __builtin_convertvector(tacc[mt][1], f8v) * xv;
        }
    }

    // Epilogue: C/D layout => VGPR v holds M = v + 8*kh, N = l15.
    #pragma unroll
    for (int mt = 0; mt < 4; ++mt) {
        #pragma unroll
        for (int v = 0; v < 8; ++v) {
            const size_t row = (size_t)(bbase + mwbase + mt * 16 + 8 * kh + v) * OUTD;
            atomicAdd(out + row + ngbase + l15,      acc[mt][0][v]);
            atomicAdd(out + row + ngbase + 16 + l15, acc[mt][1][v]);
        }
    }
}

extern "C" void kernel_launch(void* const* d_in, const int* in_sizes, int n_in,
                              void* d_out, int out_size, void* d_ws, size_t ws_size,
                              hipStream_t stream) {
    (void)in_sizes; (void)n_in; (void)out_size; (void)d_ws; (void)ws_size;
    const float* x    = (const float*)d_in[0];
    const float* z    = (const float*)d_in[1];
    const float* W    = (const float*)d_in[2];
    const float* U    = (const float*)d_in[3];
    const float* V    = (const float*)d_in[4];
    const float* bias = (const float*)d_in[5];
    float* out = (float*)d_out;

    // 1) out = bias + z@U^T + x@V^T   (also initializes for atomics)
    dim3 ginit(BATCH, OUTD / 256, 1);
    bilinear_init_uvb<<<ginit, 256, 0, stream>>>(x, z, U, V, bias, out);

    // 2) out += bilinear term via WMMA GEMM, K split across grid.z
    dim3 gmain(BATCH / BM, OUTD / BN, KSPLIT);   // 4 x 8 x 16 = 512 WGs
    bilinear_wmma<<<gmain, 256, 0, stream>>>(x, z, W, out);
}